// FT_init_49460843381066
// MI455X (gfx1250) — compile-verified
//
#include <hip/hip_runtime.h>
#include <math.h>

// ---------------------------------------------------------------------------
// Mag/phase-swap reconstruction via matmul-DFT on CDNA5 WMMA (f32 16x16x4).
//   sub  = pixel_unshuffle(x,4)                  [8,16,128,128]
//   F    = W * sub * W          (W[k][n] = exp(-2*pi*i*k*n/128), symmetric)
//   mag  = |F| ; unit = F/|F|   (unit=(1,0) when F==0, matching angle(0)=0)
//   for each (i,j):  out_real = real( IV * (mag_i .* unit_j) * IV )
//   with IV = conj(W)/128  (two applications give the 1/N^2 of ifft2)
//   scattered with pixel_shuffle(j) into out[8,16,512,512].
// ---------------------------------------------------------------------------

typedef float v2f __attribute__((ext_vector_type(2)));
typedef float v8f __attribute__((ext_vector_type(8)));

#define SRD 130                 // padded LDS row stride (floats)

// d_ws layout, in floats
#define OFF_WRE   0             // 128*128  forward DFT, real  = cos
#define OFF_WIM   16384         // 128*128  forward DFT, imag  = -sin
#define OFF_IVRE  32768         // 128*128  conj(W)/128, real  = cos/128
#define OFF_IVIM  49152         // 128*128  conj(W)/128, imag  = sin/128
#define OFF_MAG   65536         // 8*16*128*128
#define OFF_URE   (65536 + 2097152)
#define OFF_UIM   (65536 + 2*2097152)
// total ws: 6,356,992 floats = ~25.4 MB

static __device__ __forceinline__ v8f wmma4(v2f a, v2f b, v8f c) {
    // D(16x16) += A(16x4) * B(4x16), f32. 8-arg form:
    // (neg_a, A, neg_b, B, c_mod, C, reuse_a, reuse_b)
    return __builtin_amdgcn_wmma_f32_16x16x4_f32(
        false, a, false, b, (short)0, c, false, false);
}

// A fragment: lane L holds A[tm + L%16][k0 + 2*(L/16) + {0,1}]  (row-major src)
static __device__ __forceinline__ v2f ldsA(const float* src, int srd,
                                           int tm, int k0, int lr, int lh) {
    const float* p = src + (tm + lr) * srd + k0 + 2 * lh;  // 8B aligned
    v2f a; a.x = p[0]; a.y = p[1];
    return a;
}
// B fragment: lane L holds B[k0 + 2*(L/16) + {0,1}][tn + L%16]
static __device__ __forceinline__ v2f ldB(const float* src, int srd,
                                          int tn, int k0, int lr, int lh) {
    int ka = k0 + 2 * lh;
    v2f b; b.x = src[ka * srd + tn + lr]; b.y = src[(ka + 1) * srd + tn + lr];
    return b;
}

// ---------------------------------------------------------------------------
__global__ void twiddle_init(float* __restrict__ ws) {
    int t = blockIdx.x * 256 + threadIdx.x;      // 0..16383
    int k = t >> 7, n = t & 127;
    int p = (k * n) & 127;                       // exact reduction mod 128
    float u = (float)p * (1.0f / 64.0f);         // theta / pi
    float c = cospif(u);
    float s = sinpif(u);
    ws[OFF_WRE  + t] = c;
    ws[OFF_WIM  + t] = -s;
    ws[OFF_IVRE + t] = c * (1.0f / 128.0f);
    ws[OFF_IVIM + t] = s * (1.0f / 128.0f);
}

// ---------------------------------------------------------------------------
// One WG per (b,c): pixel_unshuffle gather -> F = W*X*W -> mag/unit.
__global__ __launch_bounds__(256) void fwd_fft_magphase(
    const float* __restrict__ x, float* __restrict__ ws) {
    extern __shared__ float lds[];
    float* Xs  = lds;                // [128][SRD]
    float* Tre = lds + 128 * SRD;
    float* Tim = lds + 2 * 128 * SRD;

    const int bc = blockIdx.x;       // b*16 + c
    const int b = bc >> 4, c = bc & 15;
    const float* Wre = ws + OFF_WRE;
    const float* Wim = ws + OFF_WIM;

    // phase 0: gather sub-channel c (pixel_unshuffle) into LDS
    const int ci = c >> 2, cj = c & 3;
    const float* xb = x + (size_t)b * 512 * 512;
    for (int e = threadIdx.x; e < 16384; e += 256) {
        int row = e >> 7, col = e & 127;
        Xs[row * SRD + col] = xb[(size_t)(row * 4 + ci) * 512 + col * 4 + cj];
    }
    __syncthreads();

    const int wave = threadIdx.x >> 5;
    const int lane = threadIdx.x & 31;
    const int lr = lane & 15, lh = lane >> 4;

    // phase 1: T = X * W  (A = X in LDS, B = W in L2)
    for (int t = wave; t < 64; t += 8) {
        const int tm = (t >> 3) << 4, tn = (t & 7) << 4;
        v8f accR = {}, accI = {};
        #pragma unroll 4
        for (int k0 = 0; k0 < 128; k0 += 4) {
            v2f a  = ldsA(Xs, SRD, tm, k0, lr, lh);
            v2f br = ldB(Wre, 128, tn, k0, lr, lh);
            v2f bi = ldB(Wim, 128, tn, k0, lr, lh);
            accR = wmma4(a, br, accR);
            accI = wmma4(a, bi, accI);
        }
        #pragma unroll
        for (int v = 0; v < 8; ++v) {
            int row = tm + v + 8 * lh, col = tn + lr;
            Tre[row * SRD + col] = accR[v];
            Tim[row * SRD + col] = accI[v];
        }
    }
    __syncthreads();

    // phase 2: F = W * T ; emit mag / unit
    float* mag = ws + OFF_MAG + (size_t)bc * 16384;
    float* ure = ws + OFF_URE + (size_t)bc * 16384;
    float* uim = ws + OFF_UIM + (size_t)bc * 16384;
    for (int t = wave; t < 64; t += 8) {
        const int tm = (t >> 3) << 4, tn = (t & 7) << 4;
        v8f aRR = {}, aII = {}, aRI = {}, aIR = {};
        #pragma unroll 4
        for (int k0 = 0; k0 < 128; k0 += 4) {
            v2f ar = ldsA(Wre, 128, tm, k0, lr, lh);
            v2f ai = ldsA(Wim, 128, tm, k0, lr, lh);
            v2f br = ldB(Tre, SRD, tn, k0, lr, lh);
            v2f bi = ldB(Tim, SRD, tn, k0, lr, lh);
            aRR = wmma4(ar, br, aRR);   // Wre*Tre
            aII = wmma4(ai, bi, aII);   // Wim*Tim
            aRI = wmma4(ar, bi, aRI);   // Wre*Tim
            aIR = wmma4(ai, br, aIR);   // Wim*Tre
        }
        #pragma unroll
        for (int v = 0; v < 8; ++v) {
            int row = tm + v + 8 * lh, col = tn + lr;
            float fre = aRR[v] - aII[v];
            float fim = aRI[v] + aIR[v];
            float m = sqrtf(fre * fre + fim * fim);
            int idx = row * 128 + col;
            mag[idx] = m;
            if (m > 0.0f) { ure[idx] = fre / m; uim[idx] = fim / m; }
            else          { ure[idx] = 1.0f;    uim[idx] = 0.0f;    }
        }
    }
}

// ---------------------------------------------------------------------------
// One WG per (b,i,j): R = mag_i .* unit_j ; S = R*IV ; out = real(IV*S),
// scattered at the pixel_shuffle(j) location of out[b,i].
__global__ __launch_bounds__(256) void inv_pair(
    float* __restrict__ out, const float* __restrict__ ws) {
    extern __shared__ float lds[];
    float* Rre = lds;
    float* Rim = lds + 128 * SRD;
    float* Sre = lds + 2 * 128 * SRD;
    float* Sim = lds + 3 * 128 * SRD;

    const int blk = blockIdx.x;                 // b*256 + i*16 + j
    const int b = blk >> 8, i = (blk >> 4) & 15, j = blk & 15;
    const float* IVre = ws + OFF_IVRE;
    const float* IVim = ws + OFF_IVIM;
    const float* mag = ws + OFF_MAG + (size_t)(b * 16 + i) * 16384;
    const float* ure = ws + OFF_URE + (size_t)(b * 16 + j) * 16384;
    const float* uim = ws + OFF_UIM + (size_t)(b * 16 + j) * 16384;

    // phase 0: R = mag_i .* unit_j  into LDS
    for (int e = threadIdx.x; e < 16384; e += 256) {
        int row = e >> 7, col = e & 127;
        float m = mag[e];
        Rre[row * SRD + col] = m * ure[e];
        Rim[row * SRD + col] = m * uim[e];
    }
    __syncthreads();

    const int wave = threadIdx.x >> 5;
    const int lane = threadIdx.x & 31;
    const int lr = lane & 15, lh = lane >> 4;

    // phase 1: S = R * IV  (A = R in LDS, B = IV in L2)
    for (int t = wave; t < 64; t += 8) {
        const int tm = (t >> 3) << 4, tn = (t & 7) << 4;
        v8f aRR = {}, aII = {}, aRI = {}, aIR = {};
        #pragma unroll 4
        for (int k0 = 0; k0 < 128; k0 += 4) {
            v2f are = ldsA(Rre, SRD, tm, k0, lr, lh);
            v2f aim = ldsA(Rim, SRD, tm, k0, lr, lh);
            v2f br  = ldB(IVre, 128, tn, k0, lr, lh);
            v2f bi  = ldB(IVim, 128, tn, k0, lr, lh);
            aRR = wmma4(are, br, aRR);  // Rre*IVre
            aII = wmma4(aim, bi, aII);  // Rim*IVim
            aRI = wmma4(are, bi, aRI);  // Rre*IVim
            aIR = wmma4(aim, br, aIR);  // Rim*IVre
        }
        #pragma unroll
        for (int v = 0; v < 8; ++v) {
            int row = tm + v + 8 * lh, col = tn + lr;
            Sre[row * SRD + col] = aRR[v] - aII[v];
            Sim[row * SRD + col] = aRI[v] + aIR[v];
        }
    }
    __syncthreads();

    // phase 2: x = real(IV * S) = IVre*Sre - IVim*Sim, scatter to output
    float* ob = out + (size_t)(b * 16 + i) * 512 * 512;
    const int jr = j >> 2, jc = j & 3;
    for (int t = wave; t < 64; t += 8) {
        const int tm = (t >> 3) << 4, tn = (t & 7) << 4;
        v8f a1 = {}, a2 = {};
        #pragma unroll 4
        for (int k0 = 0; k0 < 128; k0 += 4) {
            v2f ar = ldsA(IVre, 128, tm, k0, lr, lh);
            v2f ai = ldsA(IVim, 128, tm, k0, lr, lh);
            v2f br = ldB(Sre, SRD, tn, k0, lr, lh);
            v2f bi = ldB(Sim, SRD, tn, k0, lr, lh);
            a1 = wmma4(ar, br, a1);
            a2 = wmma4(ai, bi, a2);
        }
        #pragma unroll
        for (int v = 0; v < 8; ++v) {
            int row = tm + v + 8 * lh, col = tn + lr;   // (m,n) in 128x128
            ob[(size_t)(row * 4 + jr) * 512 + col * 4 + jc] = a1[v] - a2[v];
        }
    }
}

// ---------------------------------------------------------------------------
extern "C" void kernel_launch(void* const* d_in, const int* in_sizes, int n_in,
                              void* d_out, int out_size, void* d_ws, size_t ws_size,
                              hipStream_t stream) {
    (void)in_sizes; (void)n_in; (void)out_size; (void)ws_size;
    const float* x = (const float*)d_in[0];
    float* out = (float*)d_out;
    float* ws  = (float*)d_ws;

    twiddle_init<<<64, 256, 0, stream>>>(ws);

    size_t lds_fwd = (size_t)3 * 128 * SRD * sizeof(float);   // ~195 KB
    fwd_fft_magphase<<<128, 256, lds_fwd, stream>>>(x, ws);

    size_t lds_inv = (size_t)4 * 128 * SRD * sizeof(float);   // ~260 KB
    inv_pair<<<2048, 256, lds_inv, stream>>>(out, ws);
}